// DeformablePSRoIPooling_42262478193270
// MI455X (gfx1250) — compile-verified
//
#include <hip/hip_runtime.h>
#include <cstdint>
#include <cstddef>

// Problem constants (match reference)
#define SPATIAL_SCALE 0.0625f
#define POOLED   7
#define NBINS    (POOLED * POOLED)   // 49
#define TRANS_STD 0.1f
#define BATCH    2
#define CHANNELS 256
#define FH       96
#define FW       96
#define N_ROIS   512

#if defined(__HIP_DEVICE_COMPILE__)
typedef __attribute__((address_space(1))) int  as1_int;
typedef __attribute__((address_space(3))) int  as3_int;

__device__ __forceinline__ void wait_asynccnt0() {
#if __has_builtin(__builtin_amdgcn_s_wait_asynccnt)
    __builtin_amdgcn_s_wait_asynccnt(0);
#else
    asm volatile("s_wait_asynccnt 0" ::: "memory");
#endif
}
#endif

// ---------------------------------------------------------------------------
// Kernel 1: NCHW -> NHWC transpose (LDS tiled, conflict free). Staging tile
// is filled with async global->LDS loads (ASYNCcnt tracked) when available.
// ---------------------------------------------------------------------------
__global__ __launch_bounds__(256) void nchw_to_nhwc(const float* __restrict__ src,
                                                    float* __restrict__ dst) {
    __shared__ float tile[32][33];            // +1 pad: conflict-free transpose
    const int tx = threadIdx.x;               // 0..31
    const int ty = threadIdx.y;               // 0..7
    const int p0 = blockIdx.x * 32;           // flattened pixel base (h*W+w)
    const int c0 = blockIdx.y * 32;           // channel base
    const int b  = blockIdx.z;
    const int HW = FH * FW;                   // 9216, divisible by 32

#if defined(__HIP_DEVICE_COMPILE__) && __has_builtin(__builtin_amdgcn_global_load_async_to_lds_b32)
#pragma unroll
    for (int i = 0; i < 4; ++i) {
        const int c = c0 + ty + i * 8;
        __builtin_amdgcn_global_load_async_to_lds_b32(
            (as1_int*)(src + ((size_t)(b * CHANNELS + c)) * HW + p0 + tx),
            (as3_int*)&tile[ty + i * 8][tx],
            /*offset=*/0, /*cpol=*/0);
    }
    wait_asynccnt0();
#else
#pragma unroll
    for (int i = 0; i < 4; ++i) {
        const int c = c0 + ty + i * 8;
        tile[ty + i * 8][tx] = src[((size_t)(b * CHANNELS + c)) * HW + p0 + tx];
    }
#endif
    __syncthreads();
#pragma unroll
    for (int i = 0; i < 4; ++i) {
        const int prow = ty + i * 8;
        dst[((size_t)(b * HW + p0 + prow)) * CHANNELS + c0 + tx] = tile[tx][prow];
    }
}

// ---------------------------------------------------------------------------
// Kernel 2: deformable PS-ROI pooling. One block per ROI.
// Thread partition: c4 = tid & 63 -> channel quad (float4 gathers),
//                   slot = tid >> 6 -> 4 bins in flight (wave-uniform).
// Output staged in LDS (global layout), streamed out via async LDS->global.
// ---------------------------------------------------------------------------
template <bool NHWC>
__global__ __launch_bounds__(256) void dpsroi_pool(const float* __restrict__ feat,
                                                   const float* __restrict__ rois,
                                                   const float* __restrict__ trans,
                                                   float* __restrict__ out) {
    __shared__ float out_s[CHANNELS * NBINS];   // 49 KB staging tile
    const int roi  = blockIdx.x;
    const int tid  = threadIdx.x;
    const int c4   = tid & 63;                  // channels 4*c4 .. 4*c4+3
    const int slot = tid >> 6;                  // 0..3 (uniform within a wave)

    const float* r = rois + (size_t)roi * 5;
    const int   b  = (int)r[0];
    const float x1 = rintf(r[1]) * SPATIAL_SCALE - 0.5f;
    const float y1 = rintf(r[2]) * SPATIAL_SCALE - 0.5f;
    const float x2 = (rintf(r[3]) + 1.0f) * SPATIAL_SCALE - 0.5f;
    const float y2 = (rintf(r[4]) + 1.0f) * SPATIAL_SCALE - 0.5f;
    const float rw = fmaxf(x2 - x1, 0.1f);
    const float rh = fmaxf(y2 - y1, 0.1f);
    const float bin_w = rw * (1.0f / POOLED);
    const float bin_h = rh * (1.0f / POOLED);
    const float sub_w = bin_w * 0.5f;
    const float sub_h = bin_h * 0.5f;

    const float* tr = trans + (size_t)roi * 2 * NBINS;

    for (int base = 0; base < NBINS; base += 4) {
        const int bin = base + slot;
        if (bin < NBINS) {
            // Warm the next bin group's trans values (global_prefetch_b8).
            if (bin + 4 < NBINS) {
                __builtin_prefetch(&tr[bin + 4], 0, 3);
                __builtin_prefetch(&tr[NBINS + bin + 4], 0, 3);
            }
            const int ph = bin / POOLED;
            const int pw = bin - ph * POOLED;

            const float tdx = tr[bin] * TRANS_STD;
            const float tdy = tr[NBINS + bin] * TRANS_STD;
            const float wstart = (float)pw * bin_w + x1 + tdx * rw;
            const float hstart = (float)ph * bin_h + y1 + tdy * rh;

            float a0 = 0.0f, a1 = 0.0f, a2 = 0.0f, a3 = 0.0f;
            int count = 0;
#pragma unroll
            for (int iy = 0; iy < 2; ++iy) {
#pragma unroll
                for (int ix = 0; ix < 2; ++ix) {
                    const float w = wstart + (float)ix * sub_w;
                    const float h = hstart + (float)iy * sub_h;
                    if (w < -0.5f || w > (float)FW - 0.5f ||
                        h < -0.5f || h > (float)FH - 0.5f)
                        continue;
                    const float wc  = fminf(fmaxf(w, 0.0f), (float)(FW - 1));
                    const float hc  = fminf(fmaxf(h, 0.0f), (float)(FH - 1));
                    const float x0f = floorf(wc), y0f = floorf(hc);
                    const float dx  = wc - x0f,  dy  = hc - y0f;
                    const int x0i = (int)x0f,       y0i = (int)y0f;
                    const int x1i = (int)ceilf(wc), y1i = (int)ceilf(hc);
                    const float w00 = (1.0f - dx) * (1.0f - dy);
                    const float w01 = dx * (1.0f - dy);
                    const float w10 = (1.0f - dx) * dy;
                    const float w11 = dx * dy;

                    if (NHWC) {
                        const float* fb =
                            feat + ((size_t)b * FH * FW) * CHANNELS + 4 * c4;
                        const float4 v00 = *(const float4*)(fb + (size_t)(y0i * FW + x0i) * CHANNELS);
                        const float4 v01 = *(const float4*)(fb + (size_t)(y0i * FW + x1i) * CHANNELS);
                        const float4 v10 = *(const float4*)(fb + (size_t)(y1i * FW + x0i) * CHANNELS);
                        const float4 v11 = *(const float4*)(fb + (size_t)(y1i * FW + x1i) * CHANNELS);
                        a0 += w00 * v00.x + w01 * v01.x + w10 * v10.x + w11 * v11.x;
                        a1 += w00 * v00.y + w01 * v01.y + w10 * v10.y + w11 * v11.y;
                        a2 += w00 * v00.z + w01 * v01.z + w10 * v10.z + w11 * v11.z;
                        a3 += w00 * v00.w + w01 * v01.w + w10 * v10.w + w11 * v11.w;
                    } else {
                        const float* fb =
                            feat + ((size_t)(b * CHANNELS + 4 * c4)) * FH * FW;
                        const int HW = FH * FW;
                        const int o00 = y0i * FW + x0i, o01 = y0i * FW + x1i;
                        const int o10 = y1i * FW + x0i, o11 = y1i * FW + x1i;
#pragma unroll
                        for (int j = 0; j < 4; ++j) {
                            const float* p = fb + (size_t)j * HW;
                            float& aj = (j == 0) ? a0 : (j == 1) ? a1 : (j == 2) ? a2 : a3;
                            aj += w00 * p[o00] + w01 * p[o01] + w10 * p[o10] + w11 * p[o11];
                        }
                    }
                    ++count;
                }
            }
            const float cnt = (float)count;
            out_s[(4 * c4 + 0) * NBINS + bin] = (count > 0) ? a0 / cnt : 0.0f;
            out_s[(4 * c4 + 1) * NBINS + bin] = (count > 0) ? a1 / cnt : 0.0f;
            out_s[(4 * c4 + 2) * NBINS + bin] = (count > 0) ? a2 / cnt : 0.0f;
            out_s[(4 * c4 + 3) * NBINS + bin] = (count > 0) ? a3 / cnt : 0.0f;
        }
    }
    __syncthreads();

    // Stream the ROI's contiguous 12544-float output chunk: LDS -> global.
    float* og = out + (size_t)roi * CHANNELS * NBINS;
#if defined(__HIP_DEVICE_COMPILE__) && __has_builtin(__builtin_amdgcn_global_store_async_from_lds_b32)
    for (int i = tid; i < CHANNELS * NBINS; i += 256) {
        __builtin_amdgcn_global_store_async_from_lds_b32(
            (as1_int*)(og + i),
            (as3_int*)&out_s[i],
            /*offset=*/0, /*cpol=*/0);
    }
    wait_asynccnt0();
#else
    for (int i = tid; i < CHANNELS * NBINS; i += 256)
        og[i] = out_s[i];
#endif
}

// ---------------------------------------------------------------------------
// Host launcher
// ---------------------------------------------------------------------------
extern "C" void kernel_launch(void* const* d_in, const int* in_sizes, int n_in,
                              void* d_out, int out_size, void* d_ws, size_t ws_size,
                              hipStream_t stream) {
    const float* feat  = (const float*)d_in[0];   // (2,256,96,96) f32
    const float* rois  = (const float*)d_in[1];   // (512,5) f32
    const float* trans = (const float*)d_in[2];   // (512,2,7,7) f32
    float* out = (float*)d_out;                   // (512,256,7,7) f32

    const size_t need = (size_t)BATCH * CHANNELS * FH * FW * sizeof(float); // 18.9 MB
    if (ws_size >= need) {
        float* featT = (float*)d_ws;
        dim3 grid_t((FH * FW) / 32, CHANNELS / 32, BATCH);  // (288, 8, 2)
        nchw_to_nhwc<<<grid_t, dim3(32, 8), 0, stream>>>(feat, featT);
        dpsroi_pool<true><<<dim3(N_ROIS), dim3(256), 0, stream>>>(
            featT, rois, trans, out);
    } else {
        dpsroi_pool<false><<<dim3(N_ROIS), dim3(256), 0, stream>>>(
            feat, rois, trans, out);
    }
}